// FAConv_64707977282166
// MI455X (gfx1250) — compile-verified
//
#include <hip/hip_runtime.h>
#include <cstdint>
#include <cstddef>

#define NN 100000
#define NE 1000000
#define CH 64
#define EPSV 0.1f

typedef __attribute__((ext_vector_type(2))) float v2f;
typedef __attribute__((ext_vector_type(8))) float v8f;

// ---------------------------------------------------------------------------
// K1: per-node init. One thread per node: read x row once, write EPS*x into
// out, compute sA = x.Wa, sB = x.Wb, init segmax encoding and denom.
// ---------------------------------------------------------------------------
__global__ void k_node_init(const float* __restrict__ x,
                            const float* __restrict__ W_att,
                            float* __restrict__ out,
                            float* __restrict__ sA,
                            float* __restrict__ sB,
                            int*   __restrict__ smax,
                            float* __restrict__ denom) {
    int i = blockIdx.x * blockDim.x + threadIdx.x;
    if (i >= NN) return;
    const float* xr = x + (size_t)i * CH;
    float* orow = out + (size_t)i * CH;
    float accA = 0.f, accB = 0.f;
#pragma unroll
    for (int k = 0; k < CH; k += 4) {
        float4 v  = *(const float4*)(xr + k);
        float4 wa = *(const float4*)(W_att + k);
        float4 wb = *(const float4*)(W_att + CH + k);
        accA += v.x * wa.x + v.y * wa.y + v.z * wa.z + v.w * wa.w;
        accB += v.x * wb.x + v.y * wb.y + v.z * wb.z + v.w * wb.w;
        float4 o;
        o.x = EPSV * v.x; o.y = EPSV * v.y; o.z = EPSV * v.z; o.w = EPSV * v.w;
        *(float4*)(orow + k) = o;
    }
    sA[i] = accA;
    sB[i] = accB;
    smax[i]  = __float_as_int(1.0f);  // lower bound of (score + 2.0) > 1.0
    denom[i] = 0.0f;
}

// ---------------------------------------------------------------------------
// K2: h = x @ W_msg^T via V_WMMA_F32_16X16X4_F32.
// One wave computes a 16x16 tile of h. Block = 128 threads = 4 waves, each
// wave owns one 16-column tile (CH=64 -> 4 tiles); block owns 16 rows.
// h[m][n] = sum_k x[m][k] * W_msg[n][k]  (B fragment = W row, contiguous).
// ---------------------------------------------------------------------------
__global__ void k_gemm_wmma(const float* __restrict__ x,
                            const float* __restrict__ Wmsg,
                            float* __restrict__ h) {
    const int wave  = threadIdx.x >> 5;      // 0..3 -> n tile
    const int lane  = threadIdx.x & 31;
    const int l16   = lane & 15;
    const int half  = lane >> 4;             // 0: K pair {0,1}; 1: K pair {2,3}
    const int mbase = blockIdx.x * 16;       // NN = 6250*16, exact
    const int nbase = wave * 16;
    const int mrow  = mbase + l16;           // A fragment row
    const int nrow  = nbase + l16;           // B fragment column == W_msg row

    const float* xp = x    + (size_t)mrow * CH + half * 2;
    const float* wp = Wmsg + (size_t)nrow * CH + half * 2;

    v8f acc = {};
#pragma unroll
    for (int kk = 0; kk < 16; ++kk) {        // 16 * K4 = K64
        v2f a, b;
        a.x = xp[0]; a.y = xp[1];
        b.x = wp[0]; b.y = wp[1];
        acc = __builtin_amdgcn_wmma_f32_16x16x4_f32(
            /*neg_a=*/false, a, /*neg_b=*/false, b,
            /*c_mod=*/(short)0, acc, /*reuse_a=*/false, /*reuse_b=*/false);
        xp += 4; wp += 4;
    }

    // C/D layout: VGPR r -> M = r + 8*half, N = l16 (within tile)
    const int n = nbase + l16;
#pragma unroll
    for (int r = 0; r < 8; ++r) {
        int m = mbase + half * 8 + r;
        h[(size_t)m * CH + n] = acc[r];
    }
}

// ---------------------------------------------------------------------------
// K3: per-edge score + segment max.  score in (-1,1); shift +2 so positive
// floats order as ints -> plain integer atomicMax.
// ---------------------------------------------------------------------------
__global__ void k_edge_score(const int* __restrict__ ei,
                             const float* __restrict__ sA,
                             const float* __restrict__ sB,
                             const float* __restrict__ b_att,
                             float* __restrict__ score,
                             int*   __restrict__ smax) {
    int e = blockIdx.x * blockDim.x + threadIdx.x;
    if (e >= NE) return;
    int r = ei[e];
    int c = ei[NE + e];
    float s = tanhf(sA[r] + sB[c] + b_att[0]);
    score[e] = s;
    atomicMax(smax + c, __float_as_int(s + 2.0f));
}

// ---------------------------------------------------------------------------
// K4: ex = exp(score - segmax), denom[col] += ex (hardware f32 atomic).
// ---------------------------------------------------------------------------
__global__ void k_edge_exp(const int* __restrict__ ei,
                           const int* __restrict__ smax,
                           float* __restrict__ score_ex,   // in: score, out: ex
                           float* __restrict__ denom) {
    int e = blockIdx.x * blockDim.x + threadIdx.x;
    if (e >= NE) return;
    int c = ei[NE + e];
    float m  = __int_as_float(smax[c]) - 2.0f;
    float ex = expf(score_ex[e] - m);
    score_ex[e] = ex;
    unsafeAtomicAdd(denom + c, ex);
}

// ---------------------------------------------------------------------------
// K5: scatter.  One wave per edge; lane handles 2 contiguous channels
// (float2 gather of h[row], two f32 atomic adds into out[col]).
// Contribution pre-scaled by (1-EPS); out already holds EPS*x.
// ---------------------------------------------------------------------------
__global__ void k_edge_scatter(const int* __restrict__ ei,
                               const float* __restrict__ ex,
                               const float* __restrict__ denom,
                               const float* __restrict__ h,
                               float* __restrict__ out) {
    int gid  = blockIdx.x * blockDim.x + threadIdx.x;
    int e    = gid >> 5;
    int lane = gid & 31;
    if (e >= NE) return;
    int r = ei[e];
    int c = ei[NE + e];
    float w = ex[e] / denom[c] * (1.0f - EPSV);
    const float* hp = h   + (size_t)r * CH + lane * 2;
    float*       op = out + (size_t)c * CH + lane * 2;
    float2 hv = *(const float2*)hp;
    unsafeAtomicAdd(op,     hv.x * w);
    unsafeAtomicAdd(op + 1, hv.y * w);
}

// ---------------------------------------------------------------------------
extern "C" void kernel_launch(void* const* d_in, const int* in_sizes, int n_in,
                              void* d_out, int out_size, void* d_ws, size_t ws_size,
                              hipStream_t stream) {
    const float* x     = (const float*)d_in[0];   // [NN, CH]
    const int*   ei    = (const int*)d_in[1];     // [2, NE]
    const float* W_att = (const float*)d_in[2];   // [2*CH, 1]
    const float* b_att = (const float*)d_in[3];   // [1]
    const float* W_msg = (const float*)d_in[4];   // [CH, CH]
    float* out = (float*)d_out;                   // [NN, CH]

    // Workspace layout (~31.2 MB total)
    float* h     = (float*)d_ws;                  // NN*CH
    float* sA    = h + (size_t)NN * CH;           // NN
    float* sB    = sA + NN;                       // NN
    float* denom = sB + NN;                       // NN
    int*   smax  = (int*)(denom + NN);            // NN
    float* exbuf = (float*)(smax + NN);           // NE

    // K1: node init (out = EPS*x, sA, sB, segmax/denom init)
    k_node_init<<<(NN + 255) / 256, 256, 0, stream>>>(x, W_att, out, sA, sB, smax, denom);

    // K2: h = x @ W_msg^T via WMMA (6250 row-tiles x 4 col-tiles, 1 tile/wave)
    k_gemm_wmma<<<NN / 16, 128, 0, stream>>>(x, W_msg, h);

    // K3: scores + segment max
    k_edge_score<<<(NE + 255) / 256, 256, 0, stream>>>(ei, sA, sB, b_att, exbuf, smax);

    // K4: exp + denom
    k_edge_exp<<<(NE + 255) / 256, 256, 0, stream>>>(ei, smax, exbuf, denom);

    // K5: weighted scatter-add (one wave32 per edge)
    {
        long long threads = (long long)NE * 32;
        int blocks = (int)((threads + 255) / 256);
        k_edge_scatter<<<blocks, 256, 0, stream>>>(ei, exbuf, denom, h, out);
    }
}